// D4RTLoss_77472620085624
// MI455X (gfx1250) — compile-verified
//
#include <hip/hip_runtime.h>
#include <hip/hip_bf16.h>

typedef float v2f __attribute__((ext_vector_type(2)));
typedef float v8f __attribute__((ext_vector_type(8)));

#define NBINS 512          // B*G = 8*64
#define TPB   256
#define GGRP  64

// ---------------------------------------------------------------- utilities
__device__ __forceinline__ float logtr(float x) {
  // sign(x) * log1p(|x|)
  return copysignf(log1pf(fabsf(x)), x);
}

// ---------------------------------------------------------------- kernel 0: zero group bins
__global__ void k_zero_bins(float* __restrict__ bins) {
  int i = blockIdx.x * blockDim.x + threadIdx.x;
  if (i < 3 * NBINS) bins[i] = 0.0f;
}

// ---------------------------------------------------------------- kernel 1: per-(b,g) z sums
__global__ void k_group_sums(const float* __restrict__ pp, const float* __restrict__ tp,
                             const int* __restrict__ mask, const int* __restrict__ groups,
                             float* __restrict__ bins, int total, int N) {
  __shared__ float h[3 * NBINS];
  int tid = threadIdx.x;
  for (int i = tid; i < 3 * NBINS; i += TPB) h[i] = 0.0f;
  __syncthreads();

  int stride = gridDim.x * TPB;
  for (int p = blockIdx.x * TPB + tid; p < total; p += stride) {
    if (mask[p] != 0) {
      int bin = (p / N) * GGRP + (groups[p] & (GGRP - 1));
      atomicAdd(&h[bin],             pp[3 * p + 2]);
      atomicAdd(&h[NBINS + bin],     tp[3 * p + 2]);
      atomicAdd(&h[2 * NBINS + bin], 1.0f);
    }
  }
  __syncthreads();
  for (int i = tid; i < 3 * NBINS; i += TPB)
    if (h[i] != 0.0f) atomicAdd(&bins[i], h[i]);
}

// ---------------------------------------------------------------- kernel 2: safe group means
__global__ void k_safe_means(const float* __restrict__ bins,
                             float* __restrict__ safeP, float* __restrict__ safeT) {
  int i = threadIdx.x;        // launched with 512 threads
  float cnt = bins[2 * NBINS + i];
  float mp = (cnt > 0.0f) ? bins[i]          / fmaxf(cnt, 1.0f) : 1.0f;
  float mt = (cnt > 0.0f) ? bins[NBINS + i]  / fmaxf(cnt, 1.0f) : 1.0f;
  safeP[i] = fmaxf(fabsf(mp), 1e-6f);
  safeT[i] = fmaxf(fabsf(mt), 1e-6f);
}

// ---------------------------------------------------------------- kernel 3: fused loss numerators
__global__ void k_loss(const float* __restrict__ pp,  const float* __restrict__ tp,
                       const float* __restrict__ p2,  const float* __restrict__ t2,
                       const float* __restrict__ pv,  const float* __restrict__ tv,
                       const float* __restrict__ pd,  const float* __restrict__ td,
                       const float* __restrict__ pn,  const float* __restrict__ tn,
                       const float* __restrict__ cf,
                       const int*   __restrict__ mask, const int* __restrict__ groups,
                       const float* __restrict__ safeP, const float* __restrict__ safeT,
                       float* __restrict__ partials, int total, int N) {
  float acc[8];
#pragma unroll
  for (int j = 0; j < 8; ++j) acc[j] = 0.0f;

  int tid = threadIdx.x;
  int stride = gridDim.x * TPB;
  for (int p = blockIdx.x * TPB + tid; p < total; p += stride) {
    int pnext = p + stride;
    if (pnext < total) {                      // gfx1250 global_prefetch path
      __builtin_prefetch(&pp[3 * pnext], 0, 1);
      __builtin_prefetch(&tp[3 * pnext], 0, 1);
    }
    if (mask[p] == 0) continue;

    int bin = (p / N) * GGRP + (groups[p] & (GGRP - 1));
    float sp = safeP[bin], st = safeT[bin];

    float x0 = pp[3 * p + 0], x1 = pp[3 * p + 1], x2 = pp[3 * p + 2];
    float y0 = tp[3 * p + 0], y1 = tp[3 * p + 1], y2 = tp[3 * p + 2];

    // L_3D: per-group normalized + log transform, L1
    acc[0] += fabsf(logtr(x0 / sp) - logtr(y0 / st))
            + fabsf(logtr(x1 / sp) - logtr(y1 / st))
            + fabsf(logtr(x2 / sp) - logtr(y2 / st));
    // L_raw3D (weight 0, kept for fidelity)
    acc[1] += fabsf(x0 - y0) + fabsf(x1 - y1) + fabsf(x2 - y2);
    // L_2D
    acc[2] += fabsf(p2[2 * p + 0] - t2[2 * p + 0])
            + fabsf(p2[2 * p + 1] - t2[2 * p + 1]);
    // L_vis: BCE with logits
    {
      float x = pv[p], t = tv[p];
      acc[3] += fmaxf(x, 0.0f) - x * t + log1pf(expf(-fabsf(x)));
    }
    // L_disp
    acc[4] += fabsf(pd[3 * p + 0] - td[3 * p + 0])
            + fabsf(pd[3 * p + 1] - td[3 * p + 1])
            + fabsf(pd[3 * p + 2] - td[3 * p + 2]);
    // L_normal: 1 - cos of unit vectors
    {
      float a0 = pn[3 * p + 0], a1 = pn[3 * p + 1], a2 = pn[3 * p + 2];
      float b0 = tn[3 * p + 0], b1 = tn[3 * p + 1], b2 = tn[3 * p + 2];
      float na = fmaxf(sqrtf(a0 * a0 + a1 * a1 + a2 * a2), 1e-12f);
      float nb = fmaxf(sqrtf(b0 * b0 + b1 * b1 + b2 * b2), 1e-12f);
      float cosv = (a0 * b0 + a1 * b1 + a2 * b2) / (na * nb);
      acc[5] += 1.0f - cosv;
    }
    // L_conf
    acc[6] += cf[p];
    // valid count
    acc[7] += 1.0f;
  }

  // wave32 tree reduction of 8 accumulators
#pragma unroll
  for (int j = 0; j < 8; ++j)
    for (int o = 16; o > 0; o >>= 1)
      acc[j] += __shfl_down(acc[j], o, 32);

  __shared__ float red[8][8];   // [wave][loss]
  int wave = tid >> 5, lane = tid & 31;
  if (lane == 0) {
#pragma unroll
    for (int j = 0; j < 8; ++j) red[wave][j] = acc[j];
  }
  __syncthreads();
  if (tid < 8) {
    float s = 0.0f;
    for (int w = 0; w < (TPB / 32); ++w) s += red[w][tid];  // fixed order: deterministic
    partials[blockIdx.x * 8 + tid] = s;
  }
}

// ---------------------------------------------------------------- kernel 4: WMMA column-sum + combine
// Reduce partials[nb][8] to totals[8] with V_WMMA_F32_16X16X4_F32 (A = ones),
// then apply masked-mean denominators and loss weights. Launch with exactly 32 threads.
// Requirements: nb % 4 == 0, and 8 floats of valid memory after partials[nb*8]
// (columns 8..15 read that padding; their sums are never consumed, so any finite
// garbage there is harmless — D[m][n] only accumulates column n of B).
__global__ void k_final(const float* __restrict__ partials, int nb, float* __restrict__ out) {
  int lane = threadIdx.x;
  int col  = lane & 15;
  int half = lane >> 4;

  v2f a; a[0] = 1.0f; a[1] = 1.0f;     // 16x4 all-ones A matrix
  v8f c = {};                           // zero accumulator

  // Unconditional, uniform-control loads: no exec-mask churn around WMMA.
  const float* q0 = partials + (half    ) * 8 + col;   // rows base+half
  const float* q1 = partials + (half + 2) * 8 + col;   // rows base+2+half
  int chunks = nb >> 2;                                 // nb is a multiple of 4
  for (int ch = 0; ch < chunks; ++ch) {
    v2f b;
    b[0] = q0[ch * 32];
    b[1] = q1[ch * 32];
    // D[m][n] = sum_k B[k][n] + C[m][n]  (A == ones)
    c = __builtin_amdgcn_wmma_f32_16x16x4_f32(false, a, false, b, (short)0, c,
                                              false, false);
  }

  __shared__ float tot[16];
  if (lane < 16) tot[lane] = c[0];      // D VGPR0: lanes 0..15 hold (M=0, N=lane)
  __syncthreads();

  if (lane == 0) {
    float s3d  = tot[0], sraw = tot[1], s2d = tot[2], svis = tot[3];
    float sdis = tot[4], snrm = tot[5], scnf = tot[6], cnt  = tot[7];
    float n3 = 3.0f * cnt + 1e-6f;
    float n2 = 2.0f * cnt + 1e-6f;
    float n1 = cnt + 1e-6f;
    float loss = 1.0f * (s3d  / n3)
               + 0.0f * (sraw / n3)
               + 0.1f * (s2d  / n2)
               + 0.1f * (svis / n1)
               + 0.1f * (sdis / n3)
               + 0.5f * (snrm / n1)
               + 0.2f * (scnf / n1);
    out[0] = loss;
  }
}

// ---------------------------------------------------------------- launcher
extern "C" void kernel_launch(void* const* d_in, const int* in_sizes, int n_in,
                              void* d_out, int out_size, void* d_ws, size_t ws_size,
                              hipStream_t stream) {
  const float* pp = (const float*)d_in[0];
  const float* tp = (const float*)d_in[1];
  const float* p2 = (const float*)d_in[2];
  const float* t2 = (const float*)d_in[3];
  const float* pv = (const float*)d_in[4];
  const float* tv = (const float*)d_in[5];
  const float* pd = (const float*)d_in[6];
  const float* td = (const float*)d_in[7];
  const float* pn = (const float*)d_in[8];
  const float* tn = (const float*)d_in[9];
  const float* cf = (const float*)d_in[10];
  const int*   mk = (const int*)d_in[11];
  const int*   gr = (const int*)d_in[12];

  int total = in_sizes[11];       // B*N
  int N     = total / 8;          // B == 8 in the reference

  float* ws    = (float*)d_ws;
  float* bins  = ws;              // 3*512
  float* safeP = ws + 3 * NBINS;  // 512
  float* safeT = safeP + NBINS;   // 512
  float* part  = safeT + NBINS;   // NB3 * 8  (+8 floats padding for k_final)

  int NB3 = 1024;
  size_t headf = (size_t)(3 * NBINS + 2 * NBINS);              // floats before partials
  size_t needf = headf + (size_t)NB3 * 8 + 8;                  // +8 padding floats
  if (ws_size < needf * sizeof(float)) {
    size_t availf = ws_size / sizeof(float);
    size_t avail  = (availf > headf + 8) ? (availf - headf - 8) / 8 : 4;
    NB3 = (int)avail;
    if (NB3 > 1024) NB3 = 1024;
    NB3 &= ~3;                   // multiple of 4 for the WMMA chunk loop
    if (NB3 < 4) NB3 = 4;
  }

  k_zero_bins <<<(3 * NBINS + TPB - 1) / TPB, TPB, 0, stream>>>(bins);
  k_group_sums<<<1024, TPB, 0, stream>>>(pp, tp, mk, gr, bins, total, N);
  k_safe_means<<<1, NBINS, 0, stream>>>(bins, safeP, safeT);
  k_loss      <<<NB3, TPB, 0, stream>>>(pp, tp, p2, t2, pv, tv, pd, td, pn, tn, cf,
                                        mk, gr, safeP, safeT, part, total, N);
  k_final     <<<1, 32, 0, stream>>>(part, NB3, (float*)d_out);

  (void)n_in; (void)out_size;
}